// MultiScaleCondConv1d_71073118815046
// MI455X (gfx1250) — compile-verified
//
#include <hip/hip_runtime.h>
#include <hip/hip_bf16.h>

// ---------------------------------------------------------------------------
// MultiScaleCondConv1d for MI455X (gfx1250, wave32, WMMA + async-to-LDS)
//
// out[b,co,t] = sum_{ci,k} cw[b,co,ci,k] * x[b,ci,t+k-3] + cb[b,co]
// cw[b] = sum_e routing[b,e] * mask_e * weight[e],
// routing = sigmoid(mean(x[b]) * fc_w + fc_b)
//
// Conv mapped to V_WMMA_F32_16X16X4_F32:  D[16 c_out x 16 t] += A[16 x 4] * B[4 x 16]
// with K = flattened (c_in,k), size 112 = 28 chunks of 4.
// x tile staged into LDS with GLOBAL_LOAD_ASYNC_TO_LDS_B128 (ASYNCcnt).
// ---------------------------------------------------------------------------

typedef __attribute__((ext_vector_type(2))) float v2f;
typedef __attribute__((ext_vector_type(8))) float v8f;

#define T_LEN   131072
#define NCH     16             // C_in == C_out
#define KMAX    7
#define KTOT    (NCH * KMAX)   // 112
#define NCHUNK  (KTOT / 4)     // 28
#define NEXP    6
#define NBATCH  16
#define BLOCK_T 512
#define XS_W    (BLOCK_T + 8)  // 520 words/row: word 0 == t0-4 (16B-aligned halo)
#define XS_STRIDE 520          // LDS row stride in words (row = 130 float4)
#define RED_BLOCKS 1024        // partial-sum blocks per batch (2048 elems each)

// ---------------- Kernel 1: deterministic partial sums of x[b] -------------
__global__ __launch_bounds__(256) void k_partial_sum(const float* __restrict__ x,
                                                     float* __restrict__ part) {
    const int b   = blockIdx.y;
    const int tid = threadIdx.x;
    const size_t base = (size_t)b * ((size_t)NCH * T_LEN) + (size_t)blockIdx.x * 2048;
    float s = 0.f;
#pragma unroll
    for (int i = 0; i < 8; ++i) s += x[base + tid + i * 256];
    __shared__ float red[256];
    red[tid] = s;
    __syncthreads();
    for (int off = 128; off > 0; off >>= 1) {
        if (tid < off) red[tid] += red[tid + off];
        __syncthreads();
    }
    if (tid == 0) part[b * RED_BLOCKS + blockIdx.x] = red[0];
}

// ---------------- Kernel 2: per-batch routing weights ----------------------
__global__ __launch_bounds__(256) void k_routing(const float* __restrict__ part,
                                                 const float* __restrict__ fc_w,
                                                 const float* __restrict__ fc_b,
                                                 float* __restrict__ rout) {
    const int b   = blockIdx.x;
    const int tid = threadIdx.x;
    float s = 0.f;
    for (int i = tid; i < RED_BLOCKS; i += 256) s += part[b * RED_BLOCKS + i];
    __shared__ float red[256];
    red[tid] = s;
    __syncthreads();
    for (int off = 128; off > 0; off >>= 1) {
        if (tid < off) red[tid] += red[tid + off];
        __syncthreads();
    }
    if (tid < NEXP) {
        float mean = red[0] * (1.0f / ((float)NCH * (float)T_LEN));
        float z = mean * fc_w[tid] + fc_b[tid];
        rout[b * NEXP + tid] = 1.0f / (1.0f + __expf(-z));
    }
}

// ---------------- Kernel 3: blend experts into WMMA-A-layout filters -------
// A-fragment layout (f32 16x16x4): lane l, vgpr v holds A[M = l%16, K = 2*(l/16)+v].
// For chunk j the global k-row is r = 4j + 2h + v, with (c_in = r/7, k = r%7).
// Stored interleaved as cwA[b][j][lane][v] so the conv kernel loads one v2f/lane.
__global__ __launch_bounds__(32) void k_combine(const float* __restrict__ weight,
                                                const float* __restrict__ bias,
                                                const float* __restrict__ rout,
                                                float* __restrict__ cwA,
                                                float* __restrict__ cbv) {
    const int b    = blockIdx.x;
    const int lane = threadIdx.x;
    const int cout = lane & 15;
    const int h    = lane >> 4;
    float r6[NEXP];
#pragma unroll
    for (int e = 0; e < NEXP; ++e) r6[e] = rout[b * NEXP + e];
    const int ks[NEXP] = {3, 3, 5, 5, 7, 7};
#pragma unroll
    for (int j = 0; j < NCHUNK; ++j) {
#pragma unroll
        for (int v = 0; v < 2; ++v) {
            const int r   = 4 * j + 2 * h + v;
            const int cin = r / 7;
            const int k   = r - cin * 7;
            float acc = 0.f;
#pragma unroll
            for (int e = 0; e < NEXP; ++e) {
                const int lo = (KMAX - ks[e]) >> 1;  // centered mask
                const float m = (k >= lo && k < lo + ks[e]) ? 1.0f : 0.0f;
                acc += r6[e] * m * weight[((e * NCH + cout) * NCH + cin) * KMAX + k];
            }
            cwA[(((size_t)b * NCHUNK + j) * 32 + lane) * 2 + v] = acc;
        }
    }
    if (h == 0) {
        float acc = 0.f;
#pragma unroll
        for (int e = 0; e < NEXP; ++e) acc += r6[e] * bias[e * NCH + cout];
        cbv[b * NCH + cout] = acc;
    }
}

// ---------------- Kernel 4: WMMA conv ------------------------------------
// Block = 1 batch x 512 time positions, 8 waves, each wave owns 64 positions
// (4 sub-tiles of 16). 28 x v_wmma_f32_16x16x4_f32 per 16x16 output tile.
__global__ __launch_bounds__(256) void k_condconv_wmma(const float* __restrict__ x,
                                                       const float* __restrict__ cwA,
                                                       const float* __restrict__ cbv,
                                                       float* __restrict__ out) {
    __shared__ float xs[NCH * XS_STRIDE];

    const int b   = blockIdx.y;
    const int t0  = blockIdx.x * BLOCK_T;
    const int tid = threadIdx.x;
    const float* xb = x + (size_t)b * NCH * T_LEN;

    // Stage x[b][:, t0-4 .. t0+516) into LDS. Interior blocks (uniform per
    // block -> no divergence) use GLOBAL_LOAD_ASYNC_TO_LDS_B128: 16B-aligned
    // global reads DMA'd straight into LDS, tracked by ASYNCcnt.
    const bool interior = (t0 >= 4) && (t0 + BLOCK_T + 4 <= T_LEN);
    if (interior) {
        for (int idx = tid; idx < NCH * (XS_W / 4); idx += 256) {
            const int c = idx / (XS_W / 4);
            const int q = idx - c * (XS_W / 4);
            unsigned lds_b = (unsigned)(size_t)&xs[c * XS_STRIDE + 4 * q];
            const float* gp = xb + (size_t)c * T_LEN + (t0 - 4) + 4 * q;
            asm volatile("global_load_async_to_lds_b128 %0, %1, off"
                         :: "v"(lds_b), "v"(gp) : "memory");
        }
        asm volatile("s_wait_asynccnt 0x0" ::: "memory");
    } else {
        for (int idx = tid; idx < NCH * XS_W; idx += 256) {
            const int c  = idx / XS_W;
            const int tt = idx - c * XS_W;
            const int gt = t0 - 4 + tt;
            float v = 0.f;
            if (gt >= 0 && gt < T_LEN) v = xb[(size_t)c * T_LEN + gt];
            xs[c * XS_STRIDE + tt] = v;
        }
    }
    __syncthreads();

    const int lane = tid & 31;
    const int wave = tid >> 5;
    const int col  = lane & 15;   // N (time within tile) / M select for C,D
    const int h    = lane >> 4;   // lane half

    // Per-batch filter, resident in registers for the whole block (56 VGPRs),
    // one 8-byte load per chunk per lane.
    v2f afrag[NCHUNK];
    const float* cwAb = cwA + (size_t)b * NCHUNK * 64;
#pragma unroll
    for (int j = 0; j < NCHUNK; ++j) {
        afrag[j] = *(const v2f*)(cwAb + (j * 32 + lane) * 2);
    }

    // Blended bias: D-layout M = v + 8h  ->  this lane needs cb[b][v + 8h].
    float cbl[8];
#pragma unroll
    for (int v = 0; v < 8; ++v) cbl[v] = cbv[b * NCH + v + 8 * h];

    // B-fragment LDS word offsets (time-invariant part):
    // lane l, vgpr v reads B[K = 2h+v, N = col] = x[cin(r)][t0+lt+col+k(r)-3],
    // r = 4j + 2h + v; xs word 0 corresponds to t0-4  ->  word = lt+col+k+1.
    int ldsoff[NCHUNK][2];
#pragma unroll
    for (int j = 0; j < NCHUNK; ++j) {
#pragma unroll
        for (int v = 0; v < 2; ++v) {
            const int r   = 4 * j + 2 * h + v;
            const int cin = r / 7;
            const int k   = r - cin * 7;
            ldsoff[j][v] = cin * XS_STRIDE + k + col + 1;
        }
    }

    const int wave_t = wave * 64;
#pragma unroll
    for (int sub = 0; sub < 4; ++sub) {
        const int lt = wave_t + sub * 16;   // local time base within block tile
        v8f acc;
#pragma unroll
        for (int v = 0; v < 8; ++v) acc[v] = cbl[v];  // seed with blended bias
#pragma unroll
        for (int j = 0; j < NCHUNK; ++j) {
            v2f bfrag;
            bfrag.x = xs[ldsoff[j][0] + lt];
            bfrag.y = xs[ldsoff[j][1] + lt];
            acc = __builtin_amdgcn_wmma_f32_16x16x4_f32(
                /*neg_a=*/false, afrag[j], /*neg_b=*/false, bfrag,
                /*c_mod=*/(short)0, acc, /*reuse_a=*/false, /*reuse_b=*/false);
        }
        // D layout: lane stores rows M = v + 8h at col = t0 + lt + col
        // (two 64B coalesced segments per store instruction).
        float* ob = out + ((size_t)b * NCH + 8 * h) * T_LEN + t0 + lt + col;
#pragma unroll
        for (int v = 0; v < 8; ++v) ob[(size_t)v * T_LEN] = acc[v];
    }
}

// ---------------------------------------------------------------------------
extern "C" void kernel_launch(void* const* d_in, const int* in_sizes, int n_in,
                              void* d_out, int out_size, void* d_ws, size_t ws_size,
                              hipStream_t stream) {
    const float* x      = (const float*)d_in[0];   // (16,16,131072)
    const float* weight = (const float*)d_in[1];   // (6,16,16,7)
    const float* bias   = (const float*)d_in[2];   // (6,16)
    const float* fc_w   = (const float*)d_in[3];   // (6,)
    const float* fc_b   = (const float*)d_in[4];   // (6,)
    float* out = (float*)d_out;                    // (16,16,131072)

    // Workspace layout (floats): partials | routing | cb | cwA  (~182 KB total)
    float* ws   = (float*)d_ws;
    float* part = ws;                                  // 16*1024
    float* rout = ws + NBATCH * RED_BLOCKS;            // 16*6
    float* cbv  = rout + 96;                           // 16*16
    float* cwA  = cbv + 256;                           // 16*28*2*32

    k_partial_sum<<<dim3(RED_BLOCKS, NBATCH), 256, 0, stream>>>(x, part);
    k_routing<<<NBATCH, 256, 0, stream>>>(part, fc_w, fc_b, rout);
    k_combine<<<NBATCH, 32, 0, stream>>>(weight, bias, rout, cwA, cbv);
    k_condconv_wmma<<<dim3(T_LEN / BLOCK_T, NBATCH), 256, 0, stream>>>(x, cwA, cbv, out);
}